// CtcHead_20332375179615
// MI455X (gfx1250) — compile-verified
//
#include <hip/hip_runtime.h>
#include <math.h>

#define HD 768
#define SL 4096
#define NB 4
#define NGROUP 256
#define NTYPES 8
#define NROWS (NB * NGROUP)   // 1024 distinct head rows
#define NPOS  (NB * SL)       // 16384 positions
#define NHALF (NPOS / 2)      // 8192 sep / 8192 tok

typedef __attribute__((ext_vector_type(2))) float v2f;
typedef __attribute__((ext_vector_type(8))) float v8f;

// ---------------------------------------------------------------------------
// ws layout (floats):
//   [0,            786432)  avg     (1024 x 768)
//   [786432,      1572864)  h       (1024 x 768)
//   [1572864,     1581056)  logits  (1024 x 8)
//   [1581056,     1581058)  loss accumulators {sep, tok}
// ---------------------------------------------------------------------------

__global__ void init_acc_kernel(float* __restrict__ lossAcc) {
    lossAcc[0] = 0.f;
    lossAcc[1] = 0.f;
}

// One block per (group, batch); 768 threads each own one H column.
// cell_states depends only on (batch, indicator value) -> grouped average.
__global__ void group_avg_kernel(const float* __restrict__ enc,
                                 const int* __restrict__ ind,
                                 float* __restrict__ avg) {
    __shared__ int sInd[SL];
    const int g = blockIdx.x;
    const int b = blockIdx.y;
    for (int t = threadIdx.x; t < SL; t += blockDim.x)
        sInd[t] = ind[b * SL + t];
    __syncthreads();

    const int hh = threadIdx.x;  // 0..767
    const float* base = enc + (size_t)b * SL * HD + hh;
    float acc = 0.f;
    int cnt = 0;
    for (int s = 0; s < SL; ++s) {
        if (sInd[s] == g) { acc += base[(size_t)s * HD]; ++cnt; }
    }
    avg[(size_t)(b * NGROUP + g) * HD + hh] = cnt ? acc / (float)cnt : 0.f;
}

// h = tanh(X @ Wu^T + bu), X: 1024x768, Wu: 768x768 (row-major).
// One wave per 16x16 output tile; K-loop of 192 V_WMMA_F32_16X16X4_F32.
// f32 16x4 A layout: lane = 16*half + m holds A[m][2*half + j] in VGPR j.
// f32 4x16 B layout: lane = 16*half + n holds B[2*half + j][n] in VGPR j.
// B[k][n] = Wu[n][k]  (because of the transpose), contiguous in k. 
__global__ void gemm1_tanh_kernel(const float* __restrict__ X,
                                  const float* __restrict__ Wu,
                                  const float* __restrict__ bu,
                                  float* __restrict__ Hout) {
    const int lane = threadIdx.x;      // 0..31
    const int half = lane >> 4;
    const int l16  = lane & 15;
    const int ntile = blockIdx.x * blockDim.y + threadIdx.y;  // 0..47
    const int mtile = blockIdx.y;                             // 0..63

    const float* ap = X  + (size_t)(mtile * 16 + l16) * HD + 2 * half;
    const float* bp = Wu + (size_t)(ntile * 16 + l16) * HD + 2 * half;

    v8f acc = {0.f, 0.f, 0.f, 0.f, 0.f, 0.f, 0.f, 0.f};
    for (int k = 0; k < HD; k += 4) {
        v2f a = *(const v2f*)(ap + k);
        v2f b = *(const v2f*)(bp + k);
        acc = __builtin_amdgcn_wmma_f32_16x16x4_f32(
            false, a, false, b, (short)0, acc, false, false);
    }

    const int outN = ntile * 16 + l16;
    const float bias = bu[outN];
#pragma unroll
    for (int r = 0; r < 8; ++r) {
        const int outM = mtile * 16 + r + 8 * half;  // C/D: VGPR r = rows r, r+8
        Hout[(size_t)outM * HD + outN] = tanhf(acc[r] + bias);
    }
}

// logits = h @ Wp^T + bp, Wp: 8x768. One N-tile of 16 (8 valid columns).
__global__ void gemm2_kernel(const float* __restrict__ Hin,
                             const float* __restrict__ Wp,
                             const float* __restrict__ bp,
                             float* __restrict__ logits) {
    const int lane = threadIdx.x;
    const int half = lane >> 4;
    const int l16  = lane & 15;
    const int mtile = blockIdx.x * blockDim.y + threadIdx.y;  // 0..63

    const bool valid = l16 < NTYPES;
    const float* ap = Hin + (size_t)(mtile * 16 + l16) * HD + 2 * half;
    const float* bq = Wp  + (size_t)(valid ? l16 : 0) * HD + 2 * half;
    const v2f zero = {0.f, 0.f};

    v8f acc = {0.f, 0.f, 0.f, 0.f, 0.f, 0.f, 0.f, 0.f};
    for (int k = 0; k < HD; k += 4) {
        v2f a = *(const v2f*)(ap + k);
        v2f b = *(const v2f*)(bq + k);
        b = valid ? b : zero;  // v_cndmask, no divergence around WMMA
        acc = __builtin_amdgcn_wmma_f32_16x16x4_f32(
            false, a, false, b, (short)0, acc, false, false);
    }

    if (valid) {
        const float bias = bp[l16];
#pragma unroll
        for (int r = 0; r < 8; ++r) {
            const int outM = mtile * 16 + r + 8 * half;
            logits[(size_t)outM * NTYPES + l16] = acc[r] + bias;
        }
    }
}

// Per-position: gather logits by (batch, group), log-softmax, argmax, CE.
// Even flat index -> sep stream, odd -> tok stream (labels are all >= 0).
__global__ void loss_kernel(const float* __restrict__ logits,
                            const int* __restrict__ ind,
                            const int* __restrict__ lab,
                            float* __restrict__ out,
                            float* __restrict__ lossAcc) {
    const int i = blockIdx.x * blockDim.x + threadIdx.x;  // 0..16383
    const int b = i >> 12;
    const int g = ind[i] & (NGROUP - 1);
    const int row = b * NGROUP + g;

    float l[NTYPES];
#pragma unroll
    for (int c = 0; c < NTYPES; ++c) l[c] = logits[row * NTYPES + c];

    float mx = l[0];
    int pred = 0;
#pragma unroll
    for (int c = 1; c < NTYPES; ++c)
        if (l[c] > mx) { mx = l[c]; pred = c; }  // first max, like jnp.argmax

    float sum = 0.f;
#pragma unroll
    for (int c = 0; c < NTYPES; ++c) sum += expf(l[c] - mx);
    const float lse = mx + logf(sum);

    const int label = lab[i] & (NTYPES - 1);
    const float nll = lse - l[label];

    const int ord = i >> 1;
    if ((i & 1) == 0) {
        out[1 + ord]          = (float)pred;   // sep_predict
        out[1 + NHALF + ord]  = (float)label;  // sep_labels
    } else {
        out[2 + 2 * NHALF + ord]          = (float)pred;   // tok_predict
        out[2 + 3 * NHALF + ord]          = (float)label;  // tok_labels
    }

    float sepv = (i & 1) ? 0.f : nll;
    float tokv = (i & 1) ? nll : 0.f;
#pragma unroll
    for (int off = 16; off > 0; off >>= 1) {   // wave32 reduction
        sepv += __shfl_down(sepv, off, 32);
        tokv += __shfl_down(tokv, off, 32);
    }
    if ((threadIdx.x & 31) == 0) {
        atomicAdd(&lossAcc[0], sepv);
        atomicAdd(&lossAcc[1], tokv);
    }
}

__global__ void finalize_kernel(const float* __restrict__ lossAcc,
                                float* __restrict__ out) {
    out[0]             = lossAcc[0] * (1.f / (float)NHALF);  // sep_loss
    out[1 + 2 * NHALF] = lossAcc[1] * (1.f / (float)NHALF);  // tok_loss
}

extern "C" void kernel_launch(void* const* d_in, const int* in_sizes, int n_in,
                              void* d_out, int out_size, void* d_ws, size_t ws_size,
                              hipStream_t stream) {
    const float* enc = (const float*)d_in[0];  // (4, 4096, 768) f32
    const int*   ind = (const int*)d_in[1];    // (4, 4096) int
    const int*   lab = (const int*)d_in[2];    // (4, 4096) int
    const float* Wu  = (const float*)d_in[3];  // (768, 768) f32
    const float* bu  = (const float*)d_in[4];  // (768,) f32
    const float* Wp  = (const float*)d_in[5];  // (8, 768) f32
    const float* bp  = (const float*)d_in[6];  // (8,) f32
    float* out = (float*)d_out;
    float* ws  = (float*)d_ws;

    float* avg     = ws;                    // 1024*768
    float* hbuf    = ws + 786432;           // 1024*768
    float* logits  = ws + 1572864;          // 1024*8
    float* lossAcc = ws + 1581056;          // 2

    init_acc_kernel<<<1, 1, 0, stream>>>(lossAcc);

    group_avg_kernel<<<dim3(NGROUP, NB), HD, 0, stream>>>(enc, ind, avg);

    // 48 n-tiles x 64 m-tiles, 4 waves per block
    gemm1_tanh_kernel<<<dim3(12, 64), dim3(32, 4), 0, stream>>>(avg, Wu, bu, hbuf);

    // 64 m-tiles, 4 waves per block
    gemm2_kernel<<<dim3(16), dim3(32, 4), 0, stream>>>(hbuf, Wp, bp, logits);

    loss_kernel<<<NPOS / 256, 256, 0, stream>>>(logits, ind, lab, out, lossAcc);

    finalize_kernel<<<1, 1, 0, stream>>>(lossAcc, out);
}